// GIN_39565238731025
// MI455X (gfx1250) — compile-verified
//
#include <hip/hip_runtime.h>
#include <hip/hip_bf16.h>

typedef __attribute__((ext_vector_type(2))) float v2f;
typedef __attribute__((ext_vector_type(8))) float v8f;

#define DIM      128
#define N_GRAPHS 64
#define N_VARS   8
#define WPAD     136   // LDS row stride in words: 2*136 % 64 == 16 -> disjoint banks per half-wave

// ---------------------------------------------------------------- copy s = x
__global__ void gin_copy(const float* __restrict__ x, float* __restrict__ s, int n4) {
    int i = blockIdx.x * blockDim.x + threadIdx.x;
    if (i < n4) ((float4*)s)[i] = ((const float4*)x)[i];
}

// ------------------------------------------- scatter-add: s[dst] += x[src]
// 32 threads per edge, each handles 4 consecutive floats of the 128-dim row.
__global__ void gin_scatter(const float* __restrict__ x, const int* __restrict__ src,
                            const int* __restrict__ dst, float* __restrict__ s, int nEdges) {
    long long t = (long long)blockIdx.x * blockDim.x + threadIdx.x;
    int e = (int)(t >> 5);
    if (e >= nEdges) return;
    int c = ((int)t & 31) * 4;
    int sn = src[e], dn = dst[e];
    float4 v = *(const float4*)(x + (size_t)sn * DIM + c);
    float* p = s + (size_t)dn * DIM + c;
    atomicAdd(p + 0, v.x);
    atomicAdd(p + 1, v.y);
    atomicAdd(p + 2, v.z);
    atomicAdd(p + 3, v.w);
}

// --------------------------- y = relu(s @ W + b), fp32 WMMA 16x16x4, W in LDS
// Block: 128 threads = 4 waves. Block covers 64 rows; wave w -> rows r0+16w..+15.
__global__ void __launch_bounds__(128) gin_gemm_relu(
    const float* __restrict__ s, const float* __restrict__ W,
    const float* __restrict__ bias, float* __restrict__ y, int nrows) {
    __shared__ float Wl[DIM * WPAD];   // 69,632 bytes (< 320 KB WGP LDS)

    const int tid = threadIdx.x;
    // cooperative load of W[128][128] into padded LDS, float4 granules
    {
        const float4* Wg4 = (const float4*)W;
        float4* Wl4 = (float4*)Wl;
        #pragma unroll
        for (int i = tid; i < DIM * DIM / 4; i += 128) {
            int k  = i >> 5;       // row (32 float4 per source row)
            int c4 = i & 31;       // float4 column
            Wl4[k * (WPAD / 4) + c4] = Wg4[i];
        }
    }
    __syncthreads();

    const int wave = tid >> 5;
    const int lane = tid & 31;
    const int l15  = lane & 15;
    const int hh   = lane >> 4;              // half-wave select
    const int r0   = blockIdx.x * 64 + wave * 16;

    // A-fragment rows: lane l15 owns matrix row r0+l15 (both halves same rows).
    int arow  = r0 + l15;
    int arowc = arow < nrows ? arow : (nrows - 1);   // clamp for safe loads
    const float* arp = s + (size_t)arowc * DIM + 2 * hh;

    // A fragments: v2f a[k] covers K = 4k+2h, 4k+2h+1 per ISA 16x4 f32 layout.
    v2f a[32];
    #pragma unroll
    for (int k = 0; k < 32; ++k)
        a[k] = *(const v2f*)(arp + 4 * k);

    for (int nt = 0; nt < 8; ++nt) {
        const int col = nt * 16 + l15;
        v8f acc = {};
        #pragma unroll
        for (int k = 0; k < 32; ++k) {
            int kb = 4 * k + 2 * hh;
            v2f bf;
            bf.x = Wl[kb * WPAD + col];
            bf.y = Wl[(kb + 1) * WPAD + col];
            acc = __builtin_amdgcn_wmma_f32_16x16x4_f32(
                false, a[k], false, bf, (short)0, acc, false, false);
        }
        float bv = bias[col];
        #pragma unroll
        for (int r = 0; r < 8; ++r) {
            int row = r0 + r + 8 * hh;       // D layout: VGPR r -> M = r + 8*half
            if (row < nrows) {
                float v = acc[r] + bv;
                y[(size_t)row * DIM + col] = v > 0.0f ? v : 0.0f;
            }
        }
    }
}

// ------------------------------------------------ zero pooled sums + counts
__global__ void gin_zero(float* __restrict__ p, int n) {
    for (int i = threadIdx.x; i < n; i += blockDim.x) p[i] = 0.0f;
}

// ------------------------------------- atomic per-graph sum + count pooling
__global__ void gin_pool(const float* __restrict__ h, const int* __restrict__ gid,
                         float* __restrict__ sums, float* __restrict__ cnt, int nNodes) {
    long long t = (long long)blockIdx.x * blockDim.x + threadIdx.x;
    int node = (int)(t >> 5);
    if (node >= nNodes) return;
    int c = ((int)t & 31) * 4;
    int g = gid[node];
    float4 v = *(const float4*)(h + (size_t)node * DIM + c);
    float* p = sums + g * DIM + c;
    atomicAdd(p + 0, v.x);
    atomicAdd(p + 1, v.y);
    atomicAdd(p + 2, v.z);
    atomicAdd(p + 3, v.w);
    if (c == 0) atomicAdd(cnt + g, 1.0f);
}

// ------------------------------------------- head: out = (sums/cnt) @ Wp + bp
__global__ void gin_head(const float* __restrict__ sums, const float* __restrict__ cnt,
                         const float* __restrict__ Wp, const float* __restrict__ bp,
                         float* __restrict__ out) {
    int t = threadIdx.x;               // 512 threads
    int g = t >> 3, v = t & 7;
    float c = cnt[g];
    c = c < 1.0f ? 1.0f : c;
    float acc = 0.0f;
    #pragma unroll 8
    for (int k = 0; k < DIM; ++k)
        acc += (sums[g * DIM + k] / c) * Wp[k * N_VARS + v];
    out[g * N_VARS + v] = acc + bp[v];
}

extern "C" void kernel_launch(void* const* d_in, const int* in_sizes, int n_in,
                              void* d_out, int out_size, void* d_ws, size_t ws_size,
                              hipStream_t stream) {
    const float* features = (const float*)d_in[0];
    const int*   src      = (const int*)d_in[1];
    const int*   dst      = (const int*)d_in[2];
    const int*   gid      = (const int*)d_in[3];
    const float* W1 = (const float*)d_in[4];
    const float* b1 = (const float*)d_in[5];
    const float* W2 = (const float*)d_in[6];
    const float* b2 = (const float*)d_in[7];
    const float* W3 = (const float*)d_in[8];
    const float* b3 = (const float*)d_in[9];
    const float* Wp = (const float*)d_in[10];
    const float* bp = (const float*)d_in[11];
    float* out = (float*)d_out;

    const int nNodes = in_sizes[0] / DIM;      // 100000
    const int nEdges = in_sizes[1];            // 800000

    const size_t featBytes = (size_t)nNodes * DIM * sizeof(float);
    float* bufA  = (float*)d_ws;                                   // s = x + agg
    float* bufB  = (float*)((char*)d_ws + featBytes);              // layer output h
    float* sums  = (float*)((char*)d_ws + 2 * featBytes);          // [64][128]
    float* cnt   = sums + N_GRAPHS * DIM;                          // [64]

    const int n4 = nNodes * DIM / 4;
    const int copyBlocks    = (n4 + 255) / 256;
    const int scatterBlocks = (int)(((long long)nEdges * 32 + 255) / 256);
    const int gemmBlocks    = (nNodes + 63) / 64;
    const int poolBlocks    = (int)(((long long)nNodes * 32 + 255) / 256);

    // ---- layer 1: x = features -> bufB
    gin_copy<<<copyBlocks, 256, 0, stream>>>(features, bufA, n4);
    gin_scatter<<<scatterBlocks, 256, 0, stream>>>(features, src, dst, bufA, nEdges);
    gin_gemm_relu<<<gemmBlocks, 128, 0, stream>>>(bufA, W1, b1, bufB, nNodes);

    // ---- layer 2: x = bufB -> bufB (gemm reads only bufA, so overwrite is safe)
    gin_copy<<<copyBlocks, 256, 0, stream>>>(bufB, bufA, n4);
    gin_scatter<<<scatterBlocks, 256, 0, stream>>>(bufB, src, dst, bufA, nEdges);
    gin_gemm_relu<<<gemmBlocks, 128, 0, stream>>>(bufA, W2, b2, bufB, nNodes);

    // ---- layer 3
    gin_copy<<<copyBlocks, 256, 0, stream>>>(bufB, bufA, n4);
    gin_scatter<<<scatterBlocks, 256, 0, stream>>>(bufB, src, dst, bufA, nEdges);
    gin_gemm_relu<<<gemmBlocks, 128, 0, stream>>>(bufA, W3, b3, bufB, nNodes);

    // ---- mean pooling + head
    gin_zero<<<1, 256, 0, stream>>>(sums, N_GRAPHS * DIM + N_GRAPHS);
    gin_pool<<<poolBlocks, 256, 0, stream>>>(bufB, gid, sums, cnt, nNodes);
    gin_head<<<1, N_GRAPHS * N_VARS, 0, stream>>>(sums, cnt, Wp, bp, out);
}